// LIF_layer_50912542326864
// MI455X (gfx1250) — compile-verified
//
#include <hip/hip_runtime.h>
#include <cstdint>
#include <cstddef>

// LIF scan: v = 0.5*v + x_t ; s = (v >= 1) ; v -= s   over T=6 steps.
// Memory-bound streaming kernel: stage input tiles LDS-side with the gfx1250
// async data mover (ASYNCcnt), pipeline with split s_wait_asynccnt, stream
// spikes out with non-temporal b128 stores.

typedef float v4f __attribute__((ext_vector_type(4)));

#define LIF_T 6
#define LIF_BLOCK 256

__global__ __launch_bounds__(LIF_BLOCK) void lif_scan_async_kernel(
    const float* __restrict__ in, float* __restrict__ out, unsigned nv4) {
  // 6 timestep planes of 256 float4 slots = 24 KB LDS per block.
  __shared__ v4f smem[LIF_T * LIF_BLOCK];

  const unsigned tid = blockIdx.x * LIF_BLOCK + threadIdx.x;  // float4 column id
  const bool active = (tid < nv4);

  const v4f* gin = reinterpret_cast<const v4f*>(in);

  if (active) {
    // Issue all T async global->LDS copies up front. Each lane copies its own
    // 16-byte element; a wave moves 512 B per instruction. Tracked by ASYNCcnt,
    // completes in issue order.
#pragma unroll
    for (int t = 0; t < LIF_T; ++t) {
      const v4f* src = gin + (size_t)t * nv4 + tid;
      // Low 32 bits of a generic LDS pointer == LDS byte offset (ISA 10.2).
      unsigned lds_addr =
          (unsigned)(uintptr_t)(const void*)&smem[t * LIF_BLOCK + threadIdx.x];
      asm volatile("global_load_async_to_lds_b128 %0, %1, off"
                   :
                   : "v"(lds_addr), "v"(src)
                   : "memory");
    }
  }

  v4f v = {0.0f, 0.0f, 0.0f, 0.0f};  // V_INIT = 0
  v4f* gout = reinterpret_cast<v4f*>(out);

  auto lif_step = [&](int t) {
    v4f x = smem[t * LIF_BLOCK + threadIdx.x];  // ds_load_b128
    v4f s;
#pragma unroll
    for (int c = 0; c < 4; ++c) {
      v[c] = __builtin_fmaf(v[c], 0.5f, x[c]);   // v = decay*v + I_t
      s[c] = (v[c] >= 1.0f) ? 1.0f : 0.0f;       // Heaviside(v - thr)
      v[c] -= s[c];                              // reset-by-subtraction (thr=1)
    }
    if (active) {
      // Write-once output: non-temporal b128 store, skip L2 retention.
      __builtin_nontemporal_store(s, gout + (size_t)t * nv4 + tid);
    }
  };

  // Async loads retire in order: consume plane t once ASYNCcnt <= 5-t, so
  // compute on early planes overlaps the in-flight tail loads.
  asm volatile("s_wait_asynccnt 0x5" ::: "memory");
  lif_step(0);
  asm volatile("s_wait_asynccnt 0x4" ::: "memory");
  lif_step(1);
  asm volatile("s_wait_asynccnt 0x3" ::: "memory");
  lif_step(2);
  asm volatile("s_wait_asynccnt 0x2" ::: "memory");
  lif_step(3);
  asm volatile("s_wait_asynccnt 0x1" ::: "memory");
  lif_step(4);
  asm volatile("s_wait_asynccnt 0x0" ::: "memory");
  lif_step(5);
}

extern "C" void kernel_launch(void* const* d_in, const int* in_sizes, int n_in,
                              void* d_out, int out_size, void* d_ws, size_t ws_size,
                              hipStream_t stream) {
  (void)n_in; (void)out_size; (void)d_ws; (void)ws_size;

  const float* in = (const float*)d_in[0];
  float* out = (float*)d_out;

  // in_sizes[0] = T * B * C * H * W floats; per-timestep plane = total / T.
  const unsigned total_floats = (unsigned)in_sizes[0];
  const unsigned plane_floats = total_floats / LIF_T;   // 8,388,608
  const unsigned nv4 = plane_floats / 4;                // float4 columns: 2,097,152

  dim3 block(LIF_BLOCK);
  dim3 grid((nv4 + LIF_BLOCK - 1) / LIF_BLOCK);         // 8192 blocks
  lif_scan_async_kernel<<<grid, block, 0, stream>>>(in, out, nv4);
}